// VanillaRNN_25443386262308
// MI455X (gfx1250) — compile-verified
//
#include <hip/hip_runtime.h>
#include <hip/hip_bf16.h>
#include <math.h>

typedef __attribute__((ext_vector_type(2))) float v2f;
typedef __attribute__((ext_vector_type(8))) float v8f;

#define B_DIM  4096
#define T_DIM  1024
#define VOCAB_ 32000
#define HID    16

static __device__ __forceinline__ float fast_tanh(float x) {
#if __has_builtin(__builtin_amdgcn_tanhf)
  return __builtin_amdgcn_tanhf(x);     // gfx1250 v_tanh_f32
#else
  return tanhf(x);
#endif
}

// ---------------------------------------------------------------------------
// Kernel A: embedP[v][j] = Wx_b[j] + sum_k embed[v][k] * Wx_w[j][k]
// One wave per 16-row vocab tile, 4x V_WMMA_F32_16X16X4_F32 (K = 16).
// ---------------------------------------------------------------------------
__global__ __launch_bounds__(256) void k_embed_proj(
    const float* __restrict__ embed, const float* __restrict__ Wx_w,
    const float* __restrict__ Wx_b, float* __restrict__ embedP) {
  int wave = (blockIdx.x * blockDim.x + threadIdx.x) >> 5;  // 0..1999
  int lane = threadIdx.x & 31;
  int m  = lane & 15;       // A-row / B-col / D-col index for this lane
  int hi = lane >> 4;       // 0 or 1

  float bias = Wx_b[m];     // C[m'][n]: n == (lane&15), same for all rows
  v8f c;
#pragma unroll
  for (int r = 0; r < 8; ++r) c[r] = bias;

  const float* Abase = embed + (wave * 16 + m) * HID;  // A row (vocab)
  const float* Wbase = Wx_w + m * HID;                 // B col n = m
#pragma unroll
  for (int cc = 0; cc < 4; ++cc) {
    int kb = 4 * cc + 2 * hi;
    v2f a = *(const v2f*)(Abase + kb);
    v2f b = *(const v2f*)(Wbase + kb);
    c = __builtin_amdgcn_wmma_f32_16x16x4_f32(false, a, false, b,
                                              (short)0, c, false, false);
  }
  // D element (M = r + 8*hi, N = m) -> embedP[(tile*16 + M)*16 + m]
  int rowA = wave * 16 + 8 * hi;
#pragma unroll
  for (int r = 0; r < 8; ++r)
    embedP[(rowA + r) * HID + m] = c[r];
}

// ---------------------------------------------------------------------------
// Kernel B: sequential RNN scan. 16 lanes per batch row (2 rows / wave32).
// Lane j owns h[j]; full h vector replicated per lane, refreshed each step
// with 16 intra-wave shuffles (ds_bpermute). Token stream and embedP gather
// are software-pipelined (tokens 8 steps ahead, xproj 4 steps ahead) to hide
// L2 gather latency behind the dependent tanh/dot chain.
// ---------------------------------------------------------------------------
__global__ __launch_bounds__(256) void k_rnn_scan(
    const int* __restrict__ seq, const float* __restrict__ Wh,
    const float* __restrict__ embedP, float* __restrict__ hfinal) {
  int tid = blockIdx.x * blockDim.x + threadIdx.x;
  int b = tid >> 4;          // batch row
  int j = tid & 15;          // hidden index owned by this lane
  int lane = threadIdx.x & 31;
  int rowbase = lane & 16;   // first lane of this row's half-wave

  float whr[16];             // Wh row j
#pragma unroll
  for (int k = 0; k < 16; k += 4) {
    float4 w4 = *(const float4*)(Wh + j * HID + k);
    whr[k] = w4.x; whr[k + 1] = w4.y; whr[k + 2] = w4.z; whr[k + 3] = w4.w;
  }

  float hv[16];
#pragma unroll
  for (int k = 0; k < 16; ++k) hv[k] = 0.0f;

  float hn = 0.0f;
  // One RNN step: 4-way split dot (short dep chain), tanh, 16-lane rebroadcast.
  auto step = [&](float xp) {
    float a0 = fmaf(hv[0], whr[0], xp);
    float a1 = hv[1] * whr[1];
    float a2 = hv[2] * whr[2];
    float a3 = hv[3] * whr[3];
#pragma unroll
    for (int k = 4; k < 16; k += 4) {
      a0 = fmaf(hv[k + 0], whr[k + 0], a0);
      a1 = fmaf(hv[k + 1], whr[k + 1], a1);
      a2 = fmaf(hv[k + 2], whr[k + 2], a2);
      a3 = fmaf(hv[k + 3], whr[k + 3], a3);
    }
    hn = fast_tanh((a0 + a1) + (a2 + a3));
#pragma unroll
    for (int k = 0; k < 16; ++k) hv[k] = __shfl(hn, rowbase + k, 32);
  };

  const int* srow = seq + (size_t)b * T_DIM;

  // Pipeline prologue: tokens for groups 0 and 1, xproj for group 0.
  int4 tokA = *(const int4*)(srow + 0);   // tokens t0..t0+3 (current group)
  int4 tokB = *(const int4*)(srow + 4);   // tokens for next group
  float xp0 = embedP[tokA.x * HID + j];
  float xp1 = embedP[tokA.y * HID + j];
  float xp2 = embedP[tokA.z * HID + j];
  float xp3 = embedP[tokA.w * HID + j];

  for (int t0 = 0; t0 < T_DIM; t0 += 4) {
    // Prefetch tokens two groups ahead, xproj one group ahead.
    int tn = (t0 + 8 < T_DIM) ? (t0 + 8) : 0;  // tail: dummy in-range load
    int4 tokC = *(const int4*)(srow + tn);
    float nxp0 = embedP[tokB.x * HID + j];
    float nxp1 = embedP[tokB.y * HID + j];
    float nxp2 = embedP[tokB.z * HID + j];
    float nxp3 = embedP[tokB.w * HID + j];

    step(xp0);
    step(xp1);
    step(xp2);
    step(xp3);

    xp0 = nxp0; xp1 = nxp1; xp2 = nxp2; xp3 = nxp3;
    tokB = tokC;
  }
  hfinal[b * HID + j] = hn;
}

// ---------------------------------------------------------------------------
// Kernel C: out[b][v] = hfinal[b] . out_w[v] + out_b[v]  (512 MB stream-out)
// Wave computes 16x16 tiles with 4x WMMA f32 16x16x4; A held in regs across
// the vocab loop; non-temporal coalesced stores.
// ---------------------------------------------------------------------------
__global__ __launch_bounds__(256) void k_out_gemm(
    const float* __restrict__ hfinal, const float* __restrict__ out_w,
    const float* __restrict__ out_b, float* __restrict__ out) {
  const int TILES_PER_SLOT = 25;                      // 2000 tiles / 80 slots
  int bblk = blockIdx.x;                              // 0..255 (batch block)
  int slot = blockIdx.y * 8 + (threadIdx.x >> 5);     // 0..79  (vocab slot)
  int lane = threadIdx.x & 31;
  int n  = lane & 15;
  int hi = lane >> 4;

  // A tile (16 batch rows x K=16), loaded once, reused for 25 vocab tiles
  v2f a[4];
  const float* Abase = hfinal + (bblk * 16 + n) * HID;  // m == n lane pattern
#pragma unroll
  for (int cc = 0; cc < 4; ++cc)
    a[cc] = *(const v2f*)(Abase + 4 * cc + 2 * hi);

  int rowA = bblk * 16 + 8 * hi;

  for (int i = 0; i < TILES_PER_SLOT; ++i) {
    int vt = slot * TILES_PER_SLOT + i;
    float bias = out_b[vt * 16 + n];
    v8f c;
#pragma unroll
    for (int r = 0; r < 8; ++r) c[r] = bias;

    const float* Wbase = out_w + (vt * 16 + n) * HID;
#pragma unroll
    for (int cc = 0; cc < 4; ++cc) {
      v2f bv = *(const v2f*)(Wbase + 4 * cc + 2 * hi);
      c = __builtin_amdgcn_wmma_f32_16x16x4_f32(false, a[cc], false, bv,
                                                (short)0, c, false, false);
    }
    float* obase = out + (size_t)rowA * VOCAB_ + vt * 16 + n;
#pragma unroll
    for (int r = 0; r < 8; ++r)
      __builtin_nontemporal_store(c[r], obase + (size_t)r * VOCAB_);
  }
}

// ---------------------------------------------------------------------------
extern "C" void kernel_launch(void* const* d_in, const int* in_sizes, int n_in,
                              void* d_out, int out_size, void* d_ws, size_t ws_size,
                              hipStream_t stream) {
  const int*   seq   = (const int*)  d_in[0];   // [4096,1024]
  const float* embed = (const float*)d_in[1];   // [32000,16]
  const float* Wh    = (const float*)d_in[2];   // [16,16]
  const float* Wx_w  = (const float*)d_in[3];   // [16,16]
  const float* Wx_b  = (const float*)d_in[4];   // [16]
  const float* out_w = (const float*)d_in[5];   // [32000,16]
  const float* out_b = (const float*)d_in[6];   // [32000]
  float* out = (float*)d_out;                   // [4096,32000]

  float* embedP = (float*)d_ws;                 // 32000*16 f32 = 2 MB
  float* hfinal = embedP + (size_t)VOCAB_ * HID;// 4096*16 f32 = 256 KB

  // A: fold Wx (+bias) into the embedding table once.  2000 waves.
  k_embed_proj<<<dim3(250), dim3(256), 0, stream>>>(embed, Wx_w, Wx_b, embedP);
  // B: sequential scan, 16 lanes per batch row, pipelined gather.  2048 waves.
  k_rnn_scan<<<dim3(256), dim3(256), 0, stream>>>(seq, Wh, embedP, hfinal);
  // C: output projection, WMMA tiles + 512 MB NT stream-out.  20480 waves.
  k_out_gemm<<<dim3(256, 10), dim3(256), 0, stream>>>(hfinal, out_w, out_b, out);
}